// SSIMLoss_45122926411776
// MI455X (gfx1250) — compile-verified
//
#include <hip/hip_runtime.h>
#include <math.h>

typedef __attribute__((ext_vector_type(2))) float v2f;
typedef __attribute__((ext_vector_type(8))) float v8f;

#define H_IMG 320
#define W_IMG 320
#define NPIX  (H_IMG * W_IMG)
#define NB    256          // batch
#define OH    314          // valid output height (320-7+1)
#define OW    314
#define WIN   7
#define TILE  16
#define KROWS 22           // TILE + WIN - 1
#define HSTR  17           // padded LDS row stride (bank-conflict free)
#define NMAPS 5

// ---------------------------------------------------------------------------
// Kernel 1: per-sample max of Y -> data_range[b]
// ---------------------------------------------------------------------------
__global__ __launch_bounds__(256) void ssim_range_kernel(
    const float* __restrict__ Y, float* __restrict__ drange) {
    __shared__ float red[256];
    const int b = blockIdx.x;
    const float* yb = Y + (size_t)b * NPIX;
    float m = 0.0f;
    for (int i = threadIdx.x; i < NPIX; i += 256) m = fmaxf(m, yb[i]);
    red[threadIdx.x] = m;
    __syncthreads();
#pragma unroll
    for (int off = 128; off > 0; off >>= 1) {
        if (threadIdx.x < off)
            red[threadIdx.x] = fmaxf(red[threadIdx.x], red[threadIdx.x + off]);
        __syncthreads();
    }
    if (threadIdx.x == 0) drange[b] = red[0];
}

// ---------------------------------------------------------------------------
// Kernel 2: one 16x16 SSIM output tile per wave.
//   Phase 1: sliding horizontal 7-sums of {x,y,xx,yy,xy} -> LDS (22 rows x 16)
//   Phase 2: vertical 7-sum as fp32 WMMA: Out = BandA(16x24) x H(24x16),
//            six chained v_wmma_f32_16x16x4_f32 per map (K padded with zeros)
//   Phase 3: per-pixel SSIM in registers (all 5 sums share C/D layout),
//            masked edge pixels, block tree reduction -> partial[block]
// ---------------------------------------------------------------------------
__global__ __launch_bounds__(256) void ssim_tile_kernel(
    const float* __restrict__ X, const float* __restrict__ Y,
    const float* __restrict__ drange, float* __restrict__ partial) {
    __shared__ float Hlds[8 * NMAPS * KROWS * HSTR];   // 59840 B
    __shared__ float red[256];

    const int tid  = threadIdx.x;
    const int wid  = tid >> 5;
    const int lane = tid & 31;
    const int b    = blockIdx.z;

    const int tile_r = 2 * blockIdx.y + (wid >> 2);    // 0..19
    const int tile_c = 4 * blockIdx.x + (wid & 3);     // 0..19
    const int m0 = tile_r * TILE;
    const int n0 = tile_c * TILE;

    const size_t base = (size_t)b * NPIX;
    float* Hw = &Hlds[wid * NMAPS * KROWS * HSTR];

    // ---------------- Phase 1: horizontal sliding 7-sums ----------------
    if (lane < KROWS) {
        const int rr = min(m0 + lane, H_IMG - 1);
        const float* xr = X + base + (size_t)rr * W_IMG;
        const float* yr = Y + base + (size_t)rr * W_IMG;
        float xv[KROWS], yv[KROWS];
#pragma unroll
        for (int c = 0; c < KROWS; ++c) {
            const int cc = min(n0 + c, W_IMG - 1);
            xv[c] = xr[cc];
            yv[c] = yr[cc];
        }
        float sx = 0.f, sy = 0.f, sxx = 0.f, syy = 0.f, sxy = 0.f;
#pragma unroll
        for (int k = 0; k < WIN; ++k) {
            sx  += xv[k];          sy  += yv[k];
            sxx += xv[k] * xv[k];  syy += yv[k] * yv[k];
            sxy += xv[k] * yv[k];
        }
        float* row = Hw + lane * HSTR;
#pragma unroll
        for (int c = 0; c < TILE; ++c) {
            row[0 * KROWS * HSTR + c] = sx;
            row[1 * KROWS * HSTR + c] = sy;
            row[2 * KROWS * HSTR + c] = sxx;
            row[3 * KROWS * HSTR + c] = syy;
            row[4 * KROWS * HSTR + c] = sxy;
            if (c < TILE - 1) {
                const float xo = xv[c], xn = xv[c + WIN];
                const float yo = yv[c], yn = yv[c + WIN];
                sx  += xn - xo;            sy  += yn - yo;
                sxx += xn * xn - xo * xo;  syy += yn * yn - yo * yo;
                sxy += xn * yn - xo * yo;
            }
        }
    }
    __syncthreads();

    // ---------------- Phase 2: vertical 7-sum via fp32 WMMA ----------------
    // A (16x4 chunk of 16x24 band): lanes 0-15 hold K=4s+{0,1}, lanes 16-31
    // hold K=4s+{2,3}; M = lane&15.  B (4x16 chunk): VGPR0 = rows 4s+{0|2},
    // VGPR1 = rows 4s+{1|3}, N = lane&15.  C/D: VGPR g -> M = g + 8*(lane>=16).
    const int half = lane >> 4;
    const int nn   = lane & 15;
    v8f acc[NMAPS] = {};
#pragma unroll
    for (int s = 0; s < 6; ++s) {
        const int kx0 = 4 * s + 2 * half;
        const int kx1 = kx0 + 1;
        const float ax = (kx0 >= nn && kx0 <= nn + WIN - 1) ? 1.0f : 0.0f;
        const float ay = (kx1 >= nn && kx1 <= nn + WIN - 1) ? 1.0f : 0.0f;
        v2f a; a.x = ax; a.y = ay;
        const int kr0 = min(kx0, KROWS - 1);
        const int kr1 = min(kx1, KROWS - 1);
#pragma unroll
        for (int m = 0; m < NMAPS; ++m) {
            const float b0 = (kx0 < KROWS) ? Hw[m * KROWS * HSTR + kr0 * HSTR + nn] : 0.0f;
            const float b1 = (kx1 < KROWS) ? Hw[m * KROWS * HSTR + kr1 * HSTR + nn] : 0.0f;
            v2f bb; bb.x = b0; bb.y = b1;
            acc[m] = __builtin_amdgcn_wmma_f32_16x16x4_f32(
                false, a, false, bb, (short)0, acc[m], false, false);
        }
    }

    // ---------------- Phase 3: SSIM formula + reduction ----------------
    const float inv_np   = 1.0f / 49.0f;
    const float cov_norm = 49.0f / 48.0f;
    const float dr = drange[b];
    const float C1 = (0.01f * dr) * (0.01f * dr);
    const float C2 = (0.03f * dr) * (0.03f * dr);

    float lsum = 0.0f;
#pragma unroll
    for (int g = 0; g < 8; ++g) {
        const int mo = m0 + g + 8 * half;
        const int no = n0 + nn;
        const float ux  = acc[0][g] * inv_np;
        const float uy  = acc[1][g] * inv_np;
        const float uxx = acc[2][g] * inv_np;
        const float uyy = acc[3][g] * inv_np;
        const float uxy = acc[4][g] * inv_np;
        const float vx  = cov_norm * (uxx - ux * ux);
        const float vy  = cov_norm * (uyy - uy * uy);
        const float vxy = cov_norm * (uxy - ux * uy);
        const float A1 = 2.0f * ux * uy + C1;
        const float A2 = 2.0f * vxy + C2;
        const float B1 = ux * ux + uy * uy + C1;
        const float B2 = vx + vy + C2;
        const float S  = (A1 * A2) / (B1 * B2);
        lsum += (mo < OH && no < OW) ? S : 0.0f;
    }

    red[tid] = lsum;
    __syncthreads();
#pragma unroll
    for (int off = 128; off > 0; off >>= 1) {
        if (tid < off) red[tid] += red[tid + off];
        __syncthreads();
    }
    if (tid == 0) {
        const int bid = (blockIdx.z * gridDim.y + blockIdx.y) * gridDim.x + blockIdx.x;
        partial[bid] = red[0];
    }
}

// ---------------------------------------------------------------------------
// Kernel 3: deterministic final reduction, out = 1 - mean(S)
// ---------------------------------------------------------------------------
__global__ __launch_bounds__(256) void ssim_final_kernel(
    const float* __restrict__ partial, int n, float* __restrict__ out) {
    __shared__ double red[256];
    double s = 0.0;
    for (int i = threadIdx.x; i < n; i += 256) s += (double)partial[i];
    red[threadIdx.x] = s;
    __syncthreads();
#pragma unroll
    for (int off = 128; off > 0; off >>= 1) {
        if (threadIdx.x < off) red[threadIdx.x] += red[threadIdx.x + off];
        __syncthreads();
    }
    if (threadIdx.x == 0)
        out[0] = (float)(1.0 - red[0] / ((double)NB * (double)OH * (double)OW));
}

extern "C" void kernel_launch(void* const* d_in, const int* in_sizes, int n_in,
                              void* d_out, int out_size, void* d_ws, size_t ws_size,
                              hipStream_t stream) {
    const float* X = (const float*)d_in[0];
    const float* Y = (const float*)d_in[1];
    float* out     = (float*)d_out;
    float* drange  = (float*)d_ws;        // [256]
    float* partial = drange + NB;         // [5*10*256 = 12800]

    ssim_range_kernel<<<dim3(NB), dim3(256), 0, stream>>>(Y, drange);

    dim3 grid(5, 10, NB);                 // 20x20 tiles per sample, 8 tiles/block
    ssim_tile_kernel<<<grid, dim3(256), 0, stream>>>(X, Y, drange, partial);

    const int nblocks = 5 * 10 * NB;
    ssim_final_kernel<<<dim3(1), dim3(256), 0, stream>>>(partial, nblocks, out);
}